// RLS_complex_19164144074755
// MI455X (gfx1250) — compile-verified
//
#include <hip/hip_runtime.h>

#define L_DIM 256
#define N_DIM 512
#define OUT_DIM 512

typedef __attribute__((ext_vector_type(2))) float v2f;
typedef __attribute__((ext_vector_type(8))) float v8f;

// ---------------------------------------------------------------------------
// Kernel 1: Prs[l][i] = dot(P[l][i][:], x[l][:])   (one wave per row)
// ---------------------------------------------------------------------------
__global__ __launch_bounds__(256) void prs_matvec(
    const float* __restrict__ P, const float* __restrict__ input,
    const int* __restrict__ local_idx, float* __restrict__ Prs) {
  const int gwave = (int)((blockIdx.x * blockDim.x + threadIdx.x) >> 5);
  const int lane  = (int)(threadIdx.x & 31);
  const int l = gwave >> 9;          // / 512
  const int i = gwave & (N_DIM - 1); // % 512
  const float* __restrict__ prow = P + ((size_t)l * N_DIM + i) * N_DIM;
  const float* __restrict__ xrow = input + (size_t)local_idx[l] * N_DIM;
  float acc = 0.0f;
#pragma unroll
  for (int c = 0; c < N_DIM; c += 128) {
    const float4 p = *(const float4*)(prow + c + lane * 4);
    const float4 x = *(const float4*)(xrow + c + lane * 4);
    acc += p.x * x.x + p.y * x.y + p.z * x.z + p.w * x.w;
  }
#pragma unroll
  for (int off = 16; off > 0; off >>= 1) acc += __shfl_xor(acc, off);
  if (lane == 0) Prs[gwave] = acc;
}

// ---------------------------------------------------------------------------
// Kernel 2: a[l] = 1 / (1 + dot(x[l], Prs[l]))   (one wave per l)
// ---------------------------------------------------------------------------
__global__ __launch_bounds__(32) void a_kernel(
    const float* __restrict__ Prs, const float* __restrict__ input,
    const int* __restrict__ local_idx, float* __restrict__ a) {
  const int l = (int)blockIdx.x;
  const int lane = (int)threadIdx.x;
  const float* __restrict__ xrow = input + (size_t)local_idx[l] * N_DIM;
  const float* __restrict__ prow = Prs + (size_t)l * N_DIM;
  float acc = 0.0f;
#pragma unroll
  for (int c = 0; c < N_DIM; c += 128) {
    const float4 x = *(const float4*)(xrow + c + lane * 4);
    const float4 p = *(const float4*)(prow + c + lane * 4);
    acc += x.x * p.x + x.y * p.y + x.z * p.z + x.w * p.w;
  }
#pragma unroll
  for (int off = 16; off > 0; off >>= 1) acc += __shfl_xor(acc, off);
  if (lane == 0) a[l] = 1.0f / (1.0f + acc);
}

// ---------------------------------------------------------------------------
// Kernel 3: copy w -> w_out (float4 streaming)
// ---------------------------------------------------------------------------
__global__ __launch_bounds__(256) void w_copy(const float* __restrict__ w,
                                              float* __restrict__ w_out) {
  const size_t idx = blockIdx.x * blockDim.x + threadIdx.x; // 65536 float4
  ((float4*)w_out)[idx] = ((const float4*)w)[idx];
}

// ---------------------------------------------------------------------------
// Kernel 4: w_out[local[l]][:] += -(a[l]*e[local[l]]) * Prs[l][:]
// atomic add handles (hypothetical) duplicate indices like jnp .at[].add
// ---------------------------------------------------------------------------
__global__ __launch_bounds__(256) void w_update(
    const float* __restrict__ Prs, const float* __restrict__ a,
    const float* __restrict__ error, const int* __restrict__ local_idx,
    float* __restrict__ w_out) {
  const int l = (int)blockIdx.x;
  const int j = (int)threadIdx.x;
  const int row = local_idx[l];
  const float s = -a[l] * error[row];
  atomicAdd(&w_out[(size_t)row * N_DIM + j],            s * Prs[(size_t)l * N_DIM + j]);
  atomicAdd(&w_out[(size_t)row * N_DIM + j + 256],      s * Prs[(size_t)l * N_DIM + j + 256]);
}

// ---------------------------------------------------------------------------
// Kernel 5 (dominant): P_out[l] = P[l] + (-a[l]*Prs[l]) * Prs[l]^T
// One block per l; 8 waves; each wave does 16x16 tiles via
// V_WMMA_F32_16X16X4_F32:  D = A(16x4) * B(4x16) + C(16x16)
//   A: single nonzero K-slot = -a*Prs[m]  (robust to K-slot permutation)
//   B: Prs[n] replicated into every K-slot (so any K mapping gives s_m*Prs_n)
//   C/D layout per ISA: VGPR r -> M = r + 8*(lane>=16), N = lane & 15
// ---------------------------------------------------------------------------
__global__ __launch_bounds__(256) void p_update(
    const float* __restrict__ P, const float* __restrict__ Prs,
    const float* __restrict__ a, float* __restrict__ P_out) {
  const int l    = (int)blockIdx.x;
  const int wave = (int)(threadIdx.x >> 5);
  const int lane = (int)(threadIdx.x & 31);

  const float al = a[l];
  const float* __restrict__ Pl  = P     + (size_t)l * N_DIM * N_DIM;
  float* __restrict__       Pol = P_out + (size_t)l * N_DIM * N_DIM;
  const float* __restrict__ prs = Prs   + (size_t)l * N_DIM;

  const int l15 = lane & 15;

  // 32 x 32 tiles of 16x16; wave-strided over the 1024 tiles
  for (int t = wave; t < 1024; t += 8) {
    const int tm = (t >> 5) << 4;  // tile row origin
    const int tn = (t & 31) << 4;  // tile col origin

    const float prs_m = prs[tm + l15];
    const float prs_n = prs[tn + l15];

    v2f A;
    A.x = (lane < 16) ? (-al * prs_m) : 0.0f;  // exactly one nonzero K slot
    A.y = 0.0f;
    v2f B;
    B.x = prs_n;                               // replicate across all K slots
    B.y = prs_n;

    // load 16x16 tile of P into WMMA C layout
    const int row0 = tm + ((lane >> 4) << 3);  // +0 (lanes 0-15) or +8
    const int col  = tn + l15;
    const float* __restrict__ src = Pl + (size_t)row0 * N_DIM + col;
    v8f C;
#pragma unroll
    for (int r = 0; r < 8; ++r) C[r] = src[(size_t)r * N_DIM];

    // D = A*B + C  ->  P - a * Prs Prs^T   (8 args: negA,A,negB,B,cmod,C,reuseA,reuseB)
    v8f D = __builtin_amdgcn_wmma_f32_16x16x4_f32(
        false, A, false, B, (short)0, C, false, false);

    float* __restrict__ dst = Pol + (size_t)row0 * N_DIM + col;
#pragma unroll
    for (int r = 0; r < 8; ++r) dst[(size_t)r * N_DIM] = D[r];
  }
}

// ---------------------------------------------------------------------------
// Launch
// ---------------------------------------------------------------------------
extern "C" void kernel_launch(void* const* d_in, const int* in_sizes, int n_in,
                              void* d_out, int out_size, void* d_ws, size_t ws_size,
                              hipStream_t stream) {
  const float* w       = (const float*)d_in[0];  // [512, 512]
  const float* input   = (const float*)d_in[1];  // [512, 512]
  const float* error   = (const float*)d_in[2];  // [512]
  const float* P       = (const float*)d_in[3];  // [256, 512, 512]
  const int*   local_i = (const int*)d_in[4];    // [256]

  float* out   = (float*)d_out;
  float* w_out = out;                              // 512*512 floats
  float* P_out = out + (size_t)OUT_DIM * N_DIM;    // 256*512*512 floats

  float* Prs = (float*)d_ws;                       // 256*512 floats
  float* a   = Prs + (size_t)L_DIM * N_DIM;        // 256 floats

  // 1) Prs = batched matvec (reads P once)
  prs_matvec<<<dim3((L_DIM * N_DIM) / 8), 256, 0, stream>>>(P, input, local_i, Prs);
  // 2) a[l]
  a_kernel<<<dim3(L_DIM), 32, 0, stream>>>(Prs, input, local_i, a);
  // 3) w_out = w
  w_copy<<<dim3((OUT_DIM * N_DIM) / 4 / 256), 256, 0, stream>>>(w, w_out);
  // 4) w_out[local] -= (a*e) * Prs
  w_update<<<dim3(L_DIM), 256, 0, stream>>>(Prs, a, error, local_i, w_out);
  // 5) P_out = P - a * Prs Prs^T   (WMMA rank-1 tile update, streams P once more)
  p_update<<<dim3(L_DIM), 256, 0, stream>>>(P, Prs, a, P_out);
}